// GraphSageLP_32315333935773
// MI455X (gfx1250) — compile-verified
//
#include <hip/hip_runtime.h>
#include <math.h>

typedef __attribute__((ext_vector_type(2))) float v2f;
typedef __attribute__((ext_vector_type(8))) float v8f;

#define F_IN    128
#define H_DIM   256
#define OUT_DIM 32

// ---------------------------------------------------------------------------
// CSR construction helpers
// ---------------------------------------------------------------------------
__global__ void zero_int(int* __restrict__ p, int n) {
    int i = blockIdx.x * blockDim.x + threadIdx.x;
    int stride = gridDim.x * blockDim.x;
    for (; i < n; i += stride) p[i] = 0;
}

__global__ void count_deg(const long long* __restrict__ dst,
                          int* __restrict__ cnt, long long E) {
    long long e = (long long)blockIdx.x * blockDim.x + threadIdx.x;
    if (e < E) atomicAdd(&cnt[(int)dst[e]], 1);
}

// Single-block exclusive scan (deterministic), N up to a few hundred K.
__global__ void __launch_bounds__(1024)
exclusive_scan(const int* __restrict__ cnt, int* __restrict__ off, int N) {
    __shared__ int smem[1024];
    __shared__ int carry;
    if (threadIdx.x == 0) carry = 0;
    __syncthreads();
    for (int base = 0; base < N; base += 1024) {
        int cbase = carry;
        int i = base + (int)threadIdx.x;
        int v = (i < N) ? cnt[i] : 0;
        smem[threadIdx.x] = v;
        __syncthreads();
        for (int s = 1; s < 1024; s <<= 1) {             // Hillis–Steele inclusive
            int t = (threadIdx.x >= (unsigned)s) ? smem[threadIdx.x - s] : 0;
            __syncthreads();
            smem[threadIdx.x] += t;
            __syncthreads();
        }
        if (i < N) off[i] = cbase + smem[threadIdx.x] - v;  // exclusive
        __syncthreads();
        if (threadIdx.x == 0) carry = cbase + smem[1023];
        __syncthreads();
    }
    if (threadIdx.x == 0) off[N] = carry;
}

__global__ void fill_buckets(const long long* __restrict__ src,
                             const long long* __restrict__ dst,
                             const int* __restrict__ off,
                             int* __restrict__ cursor,
                             int* __restrict__ bucket, long long E) {
    long long e = (long long)blockIdx.x * blockDim.x + threadIdx.x;
    if (e >= E) return;
    int d = (int)dst[e];
    int pos = atomicAdd(&cursor[d], 1);
    bucket[off[d] + pos] = (int)src[e];
}

// ---------------------------------------------------------------------------
// Gather-based segment max: one wave per destination node, float4 per lane.
// No atomics; fmaxf is order-independent -> deterministic output.
// ---------------------------------------------------------------------------
template <int F>
__global__ void __launch_bounds__(256)
seg_max_gather(const float* __restrict__ feat, const int* __restrict__ off,
               const int* __restrict__ bucket, float* __restrict__ agg, int N) {
    constexpr int NV = F / 128;                       // float4s per lane (1 or 2)
    int node = blockIdx.x * 8 + ((int)threadIdx.x >> 5);
    int lane = (int)threadIdx.x & 31;
    if (node >= N) return;
    int beg = off[node], end = off[node + 1];

    float4 acc[NV];
#pragma unroll
    for (int i = 0; i < NV; ++i)
        acc[i] = make_float4(-INFINITY, -INFINITY, -INFINITY, -INFINITY);

    for (int j = beg; j < end; j += 32) {
        int myidx = (j + lane < end) ? bucket[j + lane] : 0;
        int cnt = min(32, end - j);
        for (int k = 0; k < cnt; ++k) {
            int s = __shfl(myidx, k, 32);             // broadcast neighbor id
            const float4* p = (const float4*)(feat + (size_t)s * F);
#pragma unroll
            for (int i = 0; i < NV; ++i) {
                float4 v = p[lane + 32 * i];
                acc[i].x = fmaxf(acc[i].x, v.x);
                acc[i].y = fmaxf(acc[i].y, v.y);
                acc[i].z = fmaxf(acc[i].z, v.z);
                acc[i].w = fmaxf(acc[i].w, v.w);
            }
        }
    }

    float4* outp = (float4*)(agg + (size_t)node * F);
#pragma unroll
    for (int i = 0; i < NV; ++i) {
        float4 a = (beg == end) ? make_float4(0.f, 0.f, 0.f, 0.f) : acc[i];
        outp[lane + 32 * i] = a;                      // isolated nodes -> 0
    }
}

// ---------------------------------------------------------------------------
// WMMA f32 16x16x4 accumulation over one K-half.
// ---------------------------------------------------------------------------
template <int KH, int LDW>
__device__ __forceinline__ v8f wmma_accum(v8f c,
                                          const float* __restrict__ As_row,
                                          const float* __restrict__ W,
                                          int ncol, int hl, int mrow) {
#pragma unroll 8
    for (int k = 0; k < KH; k += 4) {
        v2f a, b;
        a[0] = As_row[k + 2 * hl];
        a[1] = As_row[k + 2 * hl + 1];
        b[0] = W[(size_t)(k + 2 * hl)     * LDW + ncol + mrow];
        b[1] = W[(size_t)(k + 2 * hl + 1) * LDW + ncol + mrow];
        c = __builtin_amdgcn_wmma_f32_16x16x4_f32(
                false, a, false, b, (short)0, c, false, false);
    }
    return c;
}

// ---------------------------------------------------------------------------
// Layer 1: h = relu([agg1 | x] @ [W1_l ; W1_r] + b1)   (M=16/block, N=256)
// ---------------------------------------------------------------------------
#define LDA1 260   // 256 + 4 pad; 260*4 bytes % 16 == 0 -> float4-aligned rows
__global__ void __launch_bounds__(128)
gemm1_wmma(const float* __restrict__ agg, const float* __restrict__ x,
           const float* __restrict__ Wl,  const float* __restrict__ Wr,
           const float* __restrict__ bias, float* __restrict__ h) {
    __shared__ __align__(16) float As[16 * LDA1];
    const int tid = threadIdx.x;
    const int node0 = blockIdx.x * 16;

    for (int i = tid; i < 16 * (F_IN / 4); i += 128) {   // float4 staging
        int row = i >> 5, c4 = i & 31;
        float4 a  = ((const float4*)(agg + (size_t)(node0 + row) * F_IN))[c4];
        float4 xx = ((const float4*)(x   + (size_t)(node0 + row) * F_IN))[c4];
        ((float4*)(&As[row * LDA1]))[c4]        = a;
        ((float4*)(&As[row * LDA1 + F_IN]))[c4] = xx;
    }
    __syncthreads();

    const int wave = tid >> 5, lane = tid & 31;
    const int hl = lane >> 4, mrow = lane & 15;

    for (int j = 0; j < 4; ++j) {
        const int ncol = wave * 16 + j * 64;   // covers N = 0..255
        v8f c = {0.f, 0.f, 0.f, 0.f, 0.f, 0.f, 0.f, 0.f};
        c = wmma_accum<F_IN, H_DIM>(c, &As[mrow * LDA1],        Wl, ncol, hl, mrow);
        c = wmma_accum<F_IN, H_DIM>(c, &As[mrow * LDA1 + F_IN], Wr, ncol, hl, mrow);
        const float bn = bias[ncol + mrow];
#pragma unroll
        for (int r = 0; r < 8; ++r) {
            int m = r + 8 * hl;
            float v = c[r] + bn;
            h[(size_t)(node0 + m) * H_DIM + ncol + mrow] = v > 0.f ? v : 0.f;
        }
    }
}

// ---------------------------------------------------------------------------
// Layer 2: z = [agg2 | h] @ [W2_l ; W2_r] + b2          (M=16/block, N=32)
// ---------------------------------------------------------------------------
#define LDA2 516   // 512 + 4 pad; 516*4 % 16 == 0
__global__ void __launch_bounds__(64)
gemm2_wmma(const float* __restrict__ agg, const float* __restrict__ hin,
           const float* __restrict__ Wl,  const float* __restrict__ Wr,
           const float* __restrict__ bias, float* __restrict__ z) {
    __shared__ __align__(16) float As[16 * LDA2];
    const int tid = threadIdx.x;
    const int node0 = blockIdx.x * 16;

    for (int i = tid; i < 16 * (H_DIM / 4); i += 64) {   // float4 staging
        int row = i >> 6, c4 = i & 63;
        float4 a  = ((const float4*)(agg + (size_t)(node0 + row) * H_DIM))[c4];
        float4 hh = ((const float4*)(hin + (size_t)(node0 + row) * H_DIM))[c4];
        ((float4*)(&As[row * LDA2]))[c4]         = a;
        ((float4*)(&As[row * LDA2 + H_DIM]))[c4] = hh;
    }
    __syncthreads();

    const int wave = tid >> 5, lane = tid & 31;
    const int hl = lane >> 4, mrow = lane & 15;
    const int ncol = wave * 16;                // N = 0..31

    v8f c = {0.f, 0.f, 0.f, 0.f, 0.f, 0.f, 0.f, 0.f};
    c = wmma_accum<H_DIM, OUT_DIM>(c, &As[mrow * LDA2],         Wl, ncol, hl, mrow);
    c = wmma_accum<H_DIM, OUT_DIM>(c, &As[mrow * LDA2 + H_DIM], Wr, ncol, hl, mrow);
    const float bn = bias[ncol + mrow];
#pragma unroll
    for (int r = 0; r < 8; ++r) {
        int m = r + 8 * hl;
        z[(size_t)(node0 + m) * OUT_DIM + ncol + mrow] = c[r] + bn;
    }
}

// ---------------------------------------------------------------------------
// Decode: out[e] = dot(z[s], z[d]) over 32 dims.
// ---------------------------------------------------------------------------
__global__ void decode_dot(const float* __restrict__ z,
                           const long long* __restrict__ eli,
                           float* __restrict__ out, int EL) {
    int e = blockIdx.x * blockDim.x + threadIdx.x;
    if (e >= EL) return;
    long long s = eli[e];
    long long d = eli[(size_t)EL + e];
    const float4* zs = (const float4*)(z + s * OUT_DIM);
    const float4* zd = (const float4*)(z + d * OUT_DIM);
    float acc = 0.f;
#pragma unroll
    for (int i = 0; i < 8; ++i) {
        float4 a = zs[i], b = zd[i];
        acc += a.x * b.x + a.y * b.y + a.z * b.z + a.w * b.w;
    }
    out[e] = acc;
}

// ---------------------------------------------------------------------------
extern "C" void kernel_launch(void* const* d_in, const int* in_sizes, int n_in,
                              void* d_out, int out_size, void* d_ws, size_t ws_size,
                              hipStream_t stream) {
    const float*     x   = (const float*)d_in[0];
    const long long* ei  = (const long long*)d_in[1];   // [2, E] int64
    const long long* eli = (const long long*)d_in[2];   // [2, EL] int64
    const float*     W1l = (const float*)d_in[3];
    const float*     b1  = (const float*)d_in[4];
    const float*     W1r = (const float*)d_in[5];
    const float*     W2l = (const float*)d_in[6];
    const float*     b2  = (const float*)d_in[7];
    const float*     W2r = (const float*)d_in[8];
    float*           out = (float*)d_out;

    const int       N  = in_sizes[0] / F_IN;       // 50000 (divisible by 16)
    const long long E  = in_sizes[1] / 2;          // 800000
    const int       EL = in_sizes[2] / 2;          // 200000

    float* ws   = (float*)d_ws;
    float* agg1 = ws;                               // N*128
    float* h    = agg1 + (size_t)N * F_IN;          // N*256
    float* agg2 = h    + (size_t)N * H_DIM;         // N*256
    float* z    = agg2 + (size_t)N * H_DIM;         // N*32
    int*   cnt    = (int*)(z + (size_t)N * OUT_DIM); // N  (degree, then cursor)
    int*   off    = cnt + N;                         // N+1
    int*   bucket = off + N + 1;                     // E

    const long long* esrc = ei;
    const long long* edst = ei + E;
    int eblocks = (int)((E + 255) / 256);

    // ---- Build CSR over destination nodes (reused by both layers) ----
    zero_int<<<256, 256, 0, stream>>>(cnt, N);
    count_deg<<<eblocks, 256, 0, stream>>>(edst, cnt, E);
    exclusive_scan<<<1, 1024, 0, stream>>>(cnt, off, N);
    zero_int<<<256, 256, 0, stream>>>(cnt, N);      // reuse as cursor
    fill_buckets<<<eblocks, 256, 0, stream>>>(esrc, edst, off, cnt, bucket, E);

    // ---- Layer 1 ----
    seg_max_gather<F_IN><<<(N + 7) / 8, 256, 0, stream>>>(x, off, bucket, agg1, N);
    gemm1_wmma<<<N / 16, 128, 0, stream>>>(agg1, x, W1l, W1r, b1, h);

    // ---- Layer 2 ----
    seg_max_gather<H_DIM><<<(N + 7) / 8, 256, 0, stream>>>(h, off, bucket, agg2, N);
    gemm2_wmma<<<N / 16, 64, 0, stream>>>(agg2, h, W2l, W2r, b2, z);

    // ---- Decode ----
    decode_dot<<<(EL + 255) / 256, 256, 0, stream>>>(z, eli, out, EL);
}